// GIRL_16913581212181
// MI455X (gfx1250) — compile-verified
//
#include <hip/hip_runtime.h>

typedef float v2f __attribute__((ext_vector_type(2)));
typedef float v8f __attribute__((ext_vector_type(8)));

#define DIM 128
#define LDS_PITCH 132   // 128 + 4 pad: A-fragment reads hit distinct LDS banks

// ---------------- utility kernels ----------------

__global__ void zero_kernel(float* __restrict__ p, long n) {
    long i = (long)blockIdx.x * blockDim.x + threadIdx.x;
    long stride = (long)gridDim.x * blockDim.x;
    for (; i < n; i += stride) p[i] = 0.0f;
}

__global__ void deg_kernel(const long long* __restrict__ dst, float* __restrict__ deg, int E) {
    int e = blockIdx.x * blockDim.x + threadIdx.x;
    if (e < E) unsafeAtomicAdd(&deg[(int)dst[e]], 1.0f);
}

__global__ void inv_kernel(float* __restrict__ deg, int n) {
    int i = blockIdx.x * blockDim.x + threadIdx.x;
    if (i < n) deg[i] = 1.0f / fmaxf(deg[i], 1.0f);
}

// one wave per edge: gather 512B row of feat[src], scatter-add into agg[dst]
__global__ void scatter_kernel(const float* __restrict__ feat,
                               const long long* __restrict__ src,
                               const long long* __restrict__ dst,
                               float* __restrict__ agg, int E) {
    int e = blockIdx.x * (blockDim.x >> 5) + (threadIdx.x >> 5);
    if (e >= E) return;
    int lane = threadIdx.x & 31;
    long s = (long)src[e];
    long d = (long)dst[e];
    float4 v = *reinterpret_cast<const float4*>(feat + s * DIM + lane * 4);
    float* a = agg + d * DIM + lane * 4;
    unsafeAtomicAdd(a + 0, v.x);
    unsafeAtomicAdd(a + 1, v.y);
    unsafeAtomicAdd(a + 2, v.z);
    unsafeAtomicAdd(a + 3, v.w);
}

// ---------------- fused SAGE layer: relu( (agg/deg)@Wl + x@Wr + b ) ----------------
// grid: ceil(N/16) blocks of 256 threads (8 waves). Each block: 16 rows x 128 cols of output.
// Each wave owns one 16x16 tile, chains 64 V_WMMA_F32_16X16X4_F32 (two GEMMs, K=128).

__global__ void __launch_bounds__(256)
sage_wmma_kernel(const float* __restrict__ agg, const float* __restrict__ dinv,
                 const float* __restrict__ xin,
                 const float* __restrict__ Wl, const float* __restrict__ Wr,
                 const float* __restrict__ bias, float* __restrict__ out, int Nn) {
    __shared__ float sM[16 * LDS_PITCH];
    __shared__ float sX[16 * LDS_PITCH];

    const int m0  = blockIdx.x * 16;
    const int tid = threadIdx.x;

    // cooperative stage of A-tiles into LDS: each thread handles 8 floats of mean + 8 of x
    {
        int row = tid >> 4;            // 0..15
        int cb  = (tid & 15) * 8;      // 0,8,...,120
        int gr  = m0 + row;
        float4 a0 = {0,0,0,0}, a1 = {0,0,0,0}, x0 = {0,0,0,0}, x1 = {0,0,0,0};
        float inv = 0.0f;
        if (gr < Nn) {
            inv = dinv[gr];
            const float4* ga = reinterpret_cast<const float4*>(agg + (long)gr * DIM + cb);
            const float4* gx = reinterpret_cast<const float4*>(xin + (long)gr * DIM + cb);
            a0 = ga[0]; a1 = ga[1];
            x0 = gx[0]; x1 = gx[1];
        }
        float* dm = sM + row * LDS_PITCH + cb;
        float* dx = sX + row * LDS_PITCH + cb;
        dm[0] = a0.x * inv; dm[1] = a0.y * inv; dm[2] = a0.z * inv; dm[3] = a0.w * inv;
        dm[4] = a1.x * inv; dm[5] = a1.y * inv; dm[6] = a1.z * inv; dm[7] = a1.w * inv;
        dx[0] = x0.x; dx[1] = x0.y; dx[2] = x0.z; dx[3] = x0.w;
        dx[4] = x1.x; dx[5] = x1.y; dx[6] = x1.z; dx[7] = x1.w;
    }
    __syncthreads();

    const int lane = tid & 31;
    const int wave = tid >> 5;       // 0..7 -> column tile
    const int n0   = wave * 16;
    const int m    = lane & 15;      // A row within tile / B,D column index
    const int kg   = lane >> 4;      // k-group (0 or 1)
    const int nc   = n0 + m;         // output column for this lane

    // bias depends only on column -> broadcast into all 8 accumulator rows
    float bv = bias[nc];
    v8f acc;
    #pragma unroll
    for (int r = 0; r < 8; ++r) acc[r] = bv;

    const float* sMr = sM + m * LDS_PITCH;
    const float* sXr = sX + m * LDS_PITCH;

    #pragma unroll
    for (int k = 0; k < DIM; k += 4) {
        int k0 = k + 2 * kg;         // this lane's two K values
        v2f a1v = *reinterpret_cast<const v2f*>(sMr + k0);
        v2f b1v;
        b1v[0] = Wl[(long)k0 * DIM + nc];
        b1v[1] = Wl[(long)(k0 + 1) * DIM + nc];
        acc = __builtin_amdgcn_wmma_f32_16x16x4_f32(false, a1v, false, b1v,
                                                    (short)0, acc, false, false);
        v2f a2v = *reinterpret_cast<const v2f*>(sXr + k0);
        v2f b2v;
        b2v[0] = Wr[(long)k0 * DIM + nc];
        b2v[1] = Wr[(long)(k0 + 1) * DIM + nc];
        acc = __builtin_amdgcn_wmma_f32_16x16x4_f32(false, a2v, false, b2v,
                                                    (short)0, acc, false, false);
    }

    // ReLU + store: VGPR r holds (M = r + 8*kg, N = nc)
    #pragma unroll
    for (int r = 0; r < 8; ++r) {
        int gr = m0 + r + 8 * kg;
        if (gr < Nn) {
            float v = acc[r];
            out[(long)gr * DIM + nc] = v > 0.0f ? v : 0.0f;
        }
    }
}

// ---------------- head: out = h @ W_head[128x2] + b_head ----------------

__global__ void head_kernel(const float* __restrict__ h, const float* __restrict__ W,
                            const float* __restrict__ b, float* __restrict__ out, int n) {
    int i = blockIdx.x * blockDim.x + threadIdx.x;
    if (i >= n) return;
    float a0 = b[0], a1 = b[1];
    const float4* hp = reinterpret_cast<const float4*>(h + (long)i * DIM);
    #pragma unroll 8
    for (int k4 = 0; k4 < DIM / 4; ++k4) {
        float4 v = hp[k4];
        int k = k4 * 4;
        a0 += v.x * W[(k + 0) * 2 + 0]; a1 += v.x * W[(k + 0) * 2 + 1];
        a0 += v.y * W[(k + 1) * 2 + 0]; a1 += v.y * W[(k + 1) * 2 + 1];
        a0 += v.z * W[(k + 2) * 2 + 0]; a1 += v.z * W[(k + 2) * 2 + 1];
        a0 += v.w * W[(k + 3) * 2 + 0]; a1 += v.w * W[(k + 3) * 2 + 1];
    }
    out[(long)i * 2 + 0] = a0;
    out[(long)i * 2 + 1] = a1;
}

// ---------------- launcher ----------------

extern "C" void kernel_launch(void* const* d_in, const int* in_sizes, int n_in,
                              void* d_out, int out_size, void* d_ws, size_t ws_size,
                              hipStream_t stream) {
    const float*      x   = (const float*)d_in[0];
    const long long*  ei  = (const long long*)d_in[1];
    const float*      W1l = (const float*)d_in[2];
    const float*      W1r = (const float*)d_in[3];
    const float*      b1  = (const float*)d_in[4];
    const float*      W2l = (const float*)d_in[5];
    const float*      W2r = (const float*)d_in[6];
    const float*      b2  = (const float*)d_in[7];
    const float*      Wh  = (const float*)d_in[8];
    const float*      bh  = (const float*)d_in[9];

    const int N = in_sizes[0] / DIM;
    const int E = in_sizes[1] / 2;
    const long long* src = ei;
    const long long* dst = ei + E;

    float* agg = (float*)d_ws;
    float* h1  = agg + (size_t)N * DIM;
    float* h2  = h1  + (size_t)N * DIM;
    float* deg = h2  + (size_t)N * DIM;

    const int tiles = (N + 15) / 16;

    // degree (once; graph is identical for both layers)
    zero_kernel<<<1024, 256, 0, stream>>>(deg, (long)N);
    deg_kernel<<<(E + 255) / 256, 256, 0, stream>>>(dst, deg, E);
    inv_kernel<<<(N + 255) / 256, 256, 0, stream>>>(deg, N);

    // layer 1
    zero_kernel<<<2048, 256, 0, stream>>>(agg, (long)N * DIM);
    scatter_kernel<<<(E + 7) / 8, 256, 0, stream>>>(x, src, dst, agg, E);
    sage_wmma_kernel<<<tiles, 256, 0, stream>>>(agg, deg, x, W1l, W1r, b1, h1, N);

    // layer 2
    zero_kernel<<<2048, 256, 0, stream>>>(agg, (long)N * DIM);
    scatter_kernel<<<(E + 7) / 8, 256, 0, stream>>>(h1, src, dst, agg, E);
    sage_wmma_kernel<<<tiles, 256, 0, stream>>>(agg, deg, h1, W2l, W2r, b2, h2, N);

    // head
    head_kernel<<<(N + 255) / 256, 256, 0, stream>>>(h2, Wh, bh, (float*)d_out, N);
}